// GCN_simple_31104153158271
// MI455X (gfx1250) — compile-verified
//
#include <hip/hip_runtime.h>

typedef __attribute__((ext_vector_type(2))) float v2f;
typedef __attribute__((ext_vector_type(8))) float v8f;

// ---------------------------------------------------------------------------
// Degree / normalization kernels
// ---------------------------------------------------------------------------
__global__ void deg_init_kernel(float* __restrict__ deg, int n) {
    int i = blockIdx.x * blockDim.x + threadIdx.x;
    if (i < n) deg[i] = 1.0f;  // self-loop weight
}

__global__ void deg_accum_kernel(const int* __restrict__ dst,
                                 const float* __restrict__ ew,
                                 float* __restrict__ deg, int E) {
    int e = blockIdx.x * blockDim.x + threadIdx.x;
    if (e < E) atomicAdd(&deg[dst[e]], ew[e]);
}

__global__ void dinv_kernel(float* __restrict__ deg, int n) {
    int i = blockIdx.x * blockDim.x + threadIdx.x;
    if (i < n) {
        float d = deg[i];
        deg[i] = (d > 0.0f) ? rsqrtf(d) : 0.0f;
    }
}

// ---------------------------------------------------------------------------
// f32 WMMA GEMM: C[M,N] = A[M,K] * B[K,N]
// One wave computes a 16-row strip across all N columns (N/16 accumulators),
// K-loop of V_WMMA_F32_16X16X4_F32. M must be a multiple of 16 (100000 is).
// ---------------------------------------------------------------------------
template <int K, int N>
__global__ void gemm_wmma_kernel(const float* __restrict__ A,
                                 const float* __restrict__ B,
                                 float* __restrict__ C, int Mtiles) {
    constexpr int NT = N / 16;
    const int wave = threadIdx.x >> 5;
    const int lane = threadIdx.x & 31;
    const int mtile = blockIdx.x * 4 + wave;
    if (mtile >= Mtiles) return;  // wave-uniform: EXEC stays all-ones

    const int row0  = mtile * 16;
    const int laneM = lane & 15;
    const int kSel  = (lane >> 4) * 2;  // 0 for lanes 0-15, 2 for lanes 16-31

    v8f acc[NT];
#pragma unroll
    for (int nt = 0; nt < NT; ++nt) acc[nt] = {};

    const float* aRow = A + (size_t)(row0 + laneM) * K + kSel;

    for (int k0 = 0; k0 < K; k0 += 4) {
        v2f a;
        a.x = aRow[k0];
        a.y = aRow[k0 + 1];
        const float* bPtr = B + (size_t)(k0 + kSel) * N + laneM;
#pragma unroll
        for (int nt = 0; nt < NT; ++nt) {
            v2f b;
            b.x = bPtr[nt * 16];
            b.y = bPtr[nt * 16 + N];
            acc[nt] = __builtin_amdgcn_wmma_f32_16x16x4_f32(
                false, a, false, b, (short)0, acc[nt], false, false);
        }
    }

    // C/D layout: VGPR i -> M=row0+i (lanes 0-15) / M=row0+8+i (lanes 16-31)
    const int cRow = row0 + (lane >> 4) * 8;
    const int cCol = lane & 15;
#pragma unroll
    for (int nt = 0; nt < NT; ++nt) {
#pragma unroll
        for (int i = 0; i < 8; ++i) {
            C[(size_t)(cRow + i) * N + nt * 16 + cCol] = acc[nt][i];
        }
    }
}

// ---------------------------------------------------------------------------
// Aggregation: seed accumulator with self-loop term dinv[i]^2 * h[i,f]
// ---------------------------------------------------------------------------
template <int F>
__global__ void agg_init_kernel(const float* __restrict__ h,
                                const float* __restrict__ dinv,
                                float* __restrict__ out, int total) {
    int idx = blockIdx.x * blockDim.x + threadIdx.x;
    if (idx < total) {
        int node = idx / F;
        float di = dinv[node];
        out[idx] = di * di * h[idx];
    }
}

// One wave per edge: contiguous vector gather of h[src], f32 atomics into dst
template <int F>
__global__ void agg_edges_kernel(const int* __restrict__ src,
                                 const int* __restrict__ dst,
                                 const float* __restrict__ ew,
                                 const float* __restrict__ dinv,
                                 const float* __restrict__ h,
                                 float* __restrict__ out, int E) {
    const int e = blockIdx.x * (blockDim.x >> 5) + (threadIdx.x >> 5);
    if (e >= E) return;
    const int lane = threadIdx.x & 31;
    const int s = src[e];
    const int d = dst[e];
    const float w = dinv[s] * ew[e] * dinv[d];
    constexpr int V = F / 32;  // 4 (layer1) or 2 (layer2) floats per lane
    const float* hp = h + (size_t)s * F + lane * V;
    float* op = out + (size_t)d * F + lane * V;
    float vals[V];
#pragma unroll
    for (int i = 0; i < V; ++i) vals[i] = hp[i];
#pragma unroll
    for (int i = 0; i < V; ++i) atomicAdd(op + i, w * vals[i]);
}

// act = relu(agg + b), F = 128
__global__ void bias_relu_kernel(const float* __restrict__ agg,
                                 const float* __restrict__ b,
                                 float* __restrict__ act, int total) {
    int idx = blockIdx.x * blockDim.x + threadIdx.x;
    if (idx < total) {
        float v = agg[idx] + b[idx & 127];
        act[idx] = v > 0.0f ? v : 0.0f;
    }
}

// ---------------------------------------------------------------------------
// In-place log-softmax over rows of 64 (+ bias). One wave per row.
// ---------------------------------------------------------------------------
__global__ void logsoftmax_kernel(float* __restrict__ out,
                                  const float* __restrict__ b2, int n) {
    const int row = blockIdx.x * (blockDim.x >> 5) + (threadIdx.x >> 5);
    if (row >= n) return;
    const int lane = threadIdx.x & 31;
    float* rp = out + (size_t)row * 64 + lane * 2;
    float v0 = rp[0] + b2[lane * 2];
    float v1 = rp[1] + b2[lane * 2 + 1];
    float m = fmaxf(v0, v1);
#pragma unroll
    for (int off = 16; off > 0; off >>= 1) m = fmaxf(m, __shfl_xor(m, off));
    float s = expf(v0 - m) + expf(v1 - m);
#pragma unroll
    for (int off = 16; off > 0; off >>= 1) s += __shfl_xor(s, off);
    float l = m + logf(s);
    rp[0] = v0 - l;
    rp[1] = v1 - l;
}

// ---------------------------------------------------------------------------
// Launcher
// ---------------------------------------------------------------------------
extern "C" void kernel_launch(void* const* d_in, const int* in_sizes, int n_in,
                              void* d_out, int out_size, void* d_ws, size_t ws_size,
                              hipStream_t stream) {
    const float* x   = (const float*)d_in[0];   // [N, 256]
    const int*   ei  = (const int*)d_in[1];     // [2, E]
    const float* ew  = (const float*)d_in[2];   // [E]
    const float* W1  = (const float*)d_in[3];   // [256, 128]
    const float* b1  = (const float*)d_in[4];   // [128]
    const float* W2  = (const float*)d_in[5];   // [128, 64]
    const float* b2  = (const float*)d_in[6];   // [64]

    const int N = in_sizes[0] / 256;  // 100000
    const int E = in_sizes[2];        // 1600000
    const int* src = ei;
    const int* dst = ei + E;

    // Workspace layout (floats): dinv[N] | h1[N*128] | agg1[N*128]
    float* ws   = (float*)d_ws;
    size_t dslot = ((size_t)N + 3) & ~(size_t)3;
    float* dinv = ws;
    float* h1   = ws + dslot;
    float* agg1 = h1 + (size_t)N * 128;
    float* out  = (float*)d_out;  // doubles as layer-2 accumulator [N, 64]

    const int TB = 256;

    // Degrees / symmetric norm
    deg_init_kernel<<<(N + TB - 1) / TB, TB, 0, stream>>>(dinv, N);
    deg_accum_kernel<<<(E + TB - 1) / TB, TB, 0, stream>>>(dst, ew, dinv, E);
    dinv_kernel<<<(N + TB - 1) / TB, TB, 0, stream>>>(dinv, N);

    // Layer 1: h1 = x @ W1
    const int Mtiles = N / 16;
    const int gemmBlocks = (Mtiles + 3) / 4;
    gemm_wmma_kernel<256, 128><<<gemmBlocks, 128, 0, stream>>>(x, W1, h1, Mtiles);

    // agg1 = D^-1/2 (A+I) D^-1/2 @ h1
    const int tot1 = N * 128;
    agg_init_kernel<128><<<(tot1 + TB - 1) / TB, TB, 0, stream>>>(h1, dinv, agg1, tot1);
    agg_edges_kernel<128><<<(E + 7) / 8, TB, 0, stream>>>(src, dst, ew, dinv, h1, agg1, E);

    // act = relu(agg1 + b1)  (reuse h1 buffer)
    bias_relu_kernel<<<(tot1 + TB - 1) / TB, TB, 0, stream>>>(agg1, b1, h1, tot1);

    // Layer 2: g2 = act @ W2  (reuse agg1 buffer)
    gemm_wmma_kernel<128, 64><<<gemmBlocks, 128, 0, stream>>>(h1, W2, agg1, Mtiles);

    // out = aggregate(g2), seeded with self-loop term
    const int tot2 = N * 64;
    agg_init_kernel<64><<<(tot2 + TB - 1) / TB, TB, 0, stream>>>(agg1, dinv, out, tot2);
    agg_edges_kernel<64><<<(E + 7) / 8, TB, 0, stream>>>(src, dst, ew, dinv, agg1, out, E);

    // out = log_softmax(out + b2)
    logsoftmax_kernel<<<(N + 7) / 8, TB, 0, stream>>>(out, b2, N);
}